// GNN_1_WithWater_simpler_75986561401176
// MI455X (gfx1250) — compile-verified
//
#include <hip/hip_runtime.h>

// ---------------- problem constants (match reference) ----------------
#define NLAYERS 5
#define EMB     128
#define NFEAT   40
#define NNODES  200000
#define NEDGES  400000
#define NGRAPHS 4000
#define BN_EPS  1e-5f

typedef __attribute__((ext_vector_type(2))) float v2f;
typedef __attribute__((ext_vector_type(8))) float v8f;

// ---- CDNA5 async global->LDS copy (16B per lane, ASYNCcnt-tracked) ----
__device__ __forceinline__ void async_copy16(uint32_t lds_off, const float* gptr) {
    asm volatile("global_load_async_to_lds_b128 %0, %1, off"
                 :: "v"(lds_off), "v"(gptr)
                 : "memory");
}
__device__ __forceinline__ void wait_async0() {
    asm volatile("s_wait_asynccnt 0x0" ::: "memory");
}

// ---------------- degree / normalization ----------------
__global__ void k_init_deg(float* __restrict__ deg, int n) {
    int i = blockIdx.x * blockDim.x + threadIdx.x;
    if (i < n) deg[i] = 1.0f;              // self-loop contributes 1
}

__global__ void k_deg_accum(const int* __restrict__ dst, float* __restrict__ deg, int e) {
    int i = blockIdx.x * blockDim.x + threadIdx.x;
    if (i < e) atomicAdd(&deg[dst[i]], 1.0f);
}

__global__ void k_dinv(float* __restrict__ deg, int n) {
    int i = blockIdx.x * blockDim.x + threadIdx.x;
    if (i < n) deg[i] = rsqrtf(deg[i]);    // deg >= 1 always
}

// ---------------- fp32 WMMA GEMM: C = act(A[MxK] @ B[KxNc] + bias) -----------
// Compile-time K/Nc so all global/LDS offsets are immediates.
// blockDim = 32*(Nc/16) waves; each block computes a (16*RT) x Nc output tile.
// A tiles staged into LDS with GLOBAL_LOAD_ASYNC_TO_LDS_B128 (no VGPR round
// trip); rows clamped to M-1 instead of zero-filled (garbage A row r only
// affects C row r, which is never stored when out of range). Each wave
// preloads all K/4 B fragments once into registers, then runs RT fully
// unrolled WMMA chains. LDS stride K+4: 16B-aligned, conflict-free b64 reads.
template <int K, int Nc, int RT>
__global__ void k_gemm_wmma(const float* __restrict__ A, const float* __restrict__ B,
                            const float* __restrict__ bias, float* __restrict__ C,
                            int M, int act) {
    constexpr int KP = K + 4;            // padded LDS stride (floats)
    constexpr int NK = K / 4;            // WMMA k-steps
    constexpr int ROWS = 16 * RT;        // rows per block
    __shared__ float As[ROWS * KP];

    const int tid  = threadIdx.x;
    const int lane = tid & 31;
    const int wave = tid >> 5;
    const int rowbase = blockIdx.x * ROWS;

    // ---- async A staging: 16B per lane, branch-free (clamped rows) ----
    for (int idx = tid; idx < ROWS * (K / 4); idx += blockDim.x) {
        int r  = idx / (K / 4);
        int c4 = idx - r * (K / 4);
        int row = min(rowbase + r, M - 1);
        uint32_t lds_off = (uint32_t)(uintptr_t)(&As[r * KP + 4 * c4]);
        async_copy16(lds_off, A + (size_t)row * K + 4 * c4);
    }

    const int n    = wave * 16 + (lane & 15);   // global output column
    const int koff = (lane >> 4) * 2;           // lanes 0-15: K{0,1}; 16-31: K{2,3}
    const int m    = lane & 15;                 // row within 16-row tile

    // ---- preload B fragments (immediate offsets off one base) ----
    v2f bf[NK];
    {
        const float* bp = B + (size_t)koff * Nc + n;
#pragma unroll
        for (int s = 0; s < NK; ++s) {
            bf[s].x = bp[4 * s * Nc];
            bf[s].y = bp[(4 * s + 1) * Nc];
        }
    }
    const float bv = bias ? bias[n] : 0.0f;

    wait_async0();
    __syncthreads();

    const bool full = (rowbase + ROWS) <= M;    // uniform across block

#pragma unroll
    for (int rt = 0; rt < RT; ++rt) {
        const float* as = &As[(16 * rt + m) * KP + koff];
        v8f acc = {};
#pragma unroll
        for (int s = 0; s < NK; ++s) {
            v2f a;
            a.x = as[4 * s];
            a.y = as[4 * s + 1];
            acc = __builtin_amdgcn_wmma_f32_16x16x4_f32(
                false, a, false, bf[s], (short)0, acc, false, false);
        }
        // C VGPR r -> M = r + 8*(lane/16), N = lane&15
        const int row0 = rowbase + 16 * rt + 8 * (lane >> 4);
        float* cp = C + (size_t)row0 * Nc + n;
        if (full) {
#pragma unroll
            for (int r = 0; r < 8; ++r) {
                float v = acc[r] + bv;
                if (act) v = fmaxf(v, 0.0f);
                cp[(size_t)r * Nc] = v;       // immediate offsets (Nc constexpr)
            }
        } else {
#pragma unroll
            for (int r = 0; r < 8; ++r) {
                if (row0 + r < M) {
                    float v = acc[r] + bv;
                    if (act) v = fmaxf(v, 0.0f);
                    cp[(size_t)r * Nc] = v;
                }
            }
        }
    }
}

// ---------------- GCN aggregation ----------------
// self-loop initializes agg (avoids a 102MB memset): agg = xw * dinv^2
__global__ void k_selfloop(const float4* __restrict__ xw, const float* __restrict__ dinv,
                           float4* __restrict__ agg, size_t total4) {
    size_t i = (size_t)blockIdx.x * blockDim.x + threadIdx.x;
    if (i < total4) {
        int node = (int)(i >> 5);          // (i*4) / EMB
        float d = dinv[node];
        float d2 = d * d;
        float4 v = xw[i];
        v.x *= d2; v.y *= d2; v.z *= d2; v.w *= d2;
        agg[i] = v;
    }
}

// 32 lanes per edge, each handles 4 features (coalesced 128B bursts)
__global__ void k_scatter(const int* __restrict__ src, const int* __restrict__ dst,
                          const float* __restrict__ dinv, const float* __restrict__ xw,
                          float* __restrict__ agg, int nedges) {
    size_t tid = (size_t)blockIdx.x * blockDim.x + threadIdx.x;
    int e    = (int)(tid >> 5);
    int lane = (int)(tid & 31);
    if (e < nedges) {
        int s = src[e], d = dst[e];
        float nrm = dinv[s] * dinv[d];
        const float* xs = xw  + (size_t)s * EMB;
        float*       ad = agg + (size_t)d * EMB;
#pragma unroll
        for (int j = 0; j < 4; ++j)
            atomicAdd(&ad[lane + 32 * j], xs[lane + 32 * j] * nrm);
    }
}

// ---------------- residual + bias + BN statistics (fused) ----------------
__global__ void k_finalize_bn(float* __restrict__ t, const float* __restrict__ h,
                              const float* __restrict__ bg,
                              float* __restrict__ bnsum, float* __restrict__ bnsq,
                              int n, int chunk) {
    int f = threadIdx.x;                   // 0..127
    int start = blockIdx.x * chunk;
    int end = min(start + chunk, n);
    float b = bg[f];
    float s = 0.0f, sq = 0.0f;
    for (int i = start; i < end; ++i) {
        size_t idx = (size_t)i * EMB + f;
        float v = t[idx] + b + h[idx];
        t[idx] = v;
        s += v;
        sq += v * v;
    }
    atomicAdd(&bnsum[f], s);
    atomicAdd(&bnsq[f],  sq);
}

__global__ void k_bn_params(const float* __restrict__ bnsum, const float* __restrict__ bnsq,
                            const float* __restrict__ gamma, const float* __restrict__ beta,
                            float* __restrict__ scale, float* __restrict__ shift, int n) {
    int f = threadIdx.x;
    float inv_n = 1.0f / (float)n;
    float mean = bnsum[f] * inv_n;
    float var  = bnsq[f] * inv_n - mean * mean;
    float sc = gamma[f] * rsqrtf(var + BN_EPS);
    scale[f] = sc;
    shift[f] = beta[f] - mean * sc;
}

__global__ void k_bn_apply(const float4* __restrict__ t, const float* __restrict__ scale,
                           const float* __restrict__ shift, float4* __restrict__ h,
                           size_t total4, int act) {
    size_t i = (size_t)blockIdx.x * blockDim.x + threadIdx.x;
    if (i < total4) {
        int f = (int)((i * 4) & (EMB - 1));
        float4 v = t[i];
        float4 r;
        r.x = v.x * scale[f + 0] + shift[f + 0];
        r.y = v.y * scale[f + 1] + shift[f + 1];
        r.z = v.z * scale[f + 2] + shift[f + 2];
        r.w = v.w * scale[f + 3] + shift[f + 3];
        if (act) {
            r.x = fmaxf(r.x, 0.f); r.y = fmaxf(r.y, 0.f);
            r.z = fmaxf(r.z, 0.f); r.w = fmaxf(r.w, 0.f);
        }
        h[i] = r;
    }
}

// ---------------- sum pooling (sorted batch: flush on run change) ------------
__global__ void k_pool(const float* __restrict__ h, const int* __restrict__ batch,
                       float* __restrict__ mol, int n, int chunk) {
    int f = threadIdx.x;                  // 0..127
    int start = blockIdx.x * chunk;
    int end = min(start + chunk, n);
    if (start >= end) return;
    int cur = batch[start];
    float acc = 0.0f;
    for (int i = start; i < end; ++i) {
        int b = batch[i];                 // uniform (scalar) load
        if (b != cur) {
            atomicAdd(&mol[(size_t)cur * EMB + f], acc);
            acc = 0.0f;
            cur = b;
        }
        acc += h[(size_t)i * EMB + f];
    }
    atomicAdd(&mol[(size_t)cur * EMB + f], acc);
}

// ---------------- final 64 -> 1 projection: one wave32 per graph -------------
__global__ void k_mlp3(const float* __restrict__ m2, const float* __restrict__ W3,
                       const float* __restrict__ b3, float* __restrict__ out, int g) {
    int tid = blockIdx.x * blockDim.x + threadIdx.x;
    int gi = tid >> 5;
    int lane = tid & 31;
    if (gi < g) {
        const float* r = m2 + (size_t)gi * 64;
        float s = r[lane] * W3[lane] + r[lane + 32] * W3[lane + 32];
#pragma unroll
        for (int off = 16; off > 0; off >>= 1)
            s += __shfl_xor(s, off, 32);
        if (lane == 0) out[gi] = s + b3[0];
    }
}

// ---------------- driver ----------------
extern "C" void kernel_launch(void* const* d_in, const int* in_sizes, int n_in,
                              void* d_out, int out_size, void* d_ws, size_t ws_size,
                              hipStream_t stream) {
    (void)in_sizes; (void)n_in; (void)out_size; (void)ws_size;

    const float* X     = (const float*)d_in[0];
    const int*   ei    = (const int*)  d_in[1];
    const int*   batch = (const int*)  d_in[2];
    const float* Wx    = (const float*)d_in[3];
    const float* bx    = (const float*)d_in[4];
    const float* Wg    = (const float*)d_in[5];
    const float* bg    = (const float*)d_in[6];
    const float* gamma = (const float*)d_in[7];
    const float* beta  = (const float*)d_in[8];
    const float* W1    = (const float*)d_in[9];
    const float* b1    = (const float*)d_in[10];
    const float* W2    = (const float*)d_in[11];
    const float* b2    = (const float*)d_in[12];
    const float* W3    = (const float*)d_in[13];
    const float* b3    = (const float*)d_in[14];
    float* out = (float*)d_out;

    const int* src = ei;            // edge_index[0]
    const int* dst = ei + NEDGES;   // edge_index[1]

    // workspace layout (floats)
    float* ws = (float*)d_ws;
    const size_t NH = (size_t)NNODES * EMB;
    float* h     = ws;                              // [N,128]
    float* xw    = h     + NH;                      // [N,128]
    float* agg   = xw    + NH;                      // [N,128] (also BN input t)
    float* deg   = agg   + NH;                      // [N] (becomes dinv in place)
    float* bnsum = deg   + NNODES;                  // [128]
    float* bnsq  = bnsum + EMB;                     // [128]
    float* scale = bnsq  + EMB;                     // [128]
    float* shift = scale + EMB;                     // [128]
    float* mol   = shift + EMB;                     // [G,128]
    float* m1    = mol   + (size_t)NGRAPHS * EMB;   // [G,128]
    float* m2    = m1    + (size_t)NGRAPHS * EMB;   // [G,64]

    // ---- symmetric GCN normalization ----
    k_init_deg <<<(NNODES + 255) / 256, 256, 0, stream>>>(deg, NNODES);
    k_deg_accum<<<(NEDGES + 255) / 256, 256, 0, stream>>>(dst, deg, NEDGES);
    k_dinv     <<<(NNODES + 255) / 256, 256, 0, stream>>>(deg, NNODES);

    // ---- input projection: h = X @ Wx + bx  (K=40) ----
    k_gemm_wmma<NFEAT, EMB, 4><<<(NNODES + 63) / 64, 256, 0, stream>>>(
        X, Wx, bx, h, NNODES, 0);

    // ---- GCN layers ----
    const size_t NH4 = NH / 4;
    const int vec_blocks = (int)((NH4 + 255) / 256);
    for (int l = 0; l < NLAYERS; ++l) {
        // xw = h @ Wg[l]
        k_gemm_wmma<EMB, EMB, 4><<<(NNODES + 63) / 64, 256, 0, stream>>>(
            h, Wg + (size_t)l * EMB * EMB, nullptr, xw, NNODES, 0);
        // agg = self-loop term, then edge scatter-add
        k_selfloop<<<vec_blocks, 256, 0, stream>>>(
            (const float4*)xw, deg, (float4*)agg, NH4);
        k_scatter<<<(NEDGES * 32 + 255) / 256, 256, 0, stream>>>(src, dst, deg, xw, agg, NEDGES);
        // t = agg + bg + h ; accumulate BN stats
        hipMemsetAsync(bnsum, 0, 2 * EMB * sizeof(float), stream);
        k_finalize_bn<<<(NNODES + 255) / 256, EMB, 0, stream>>>(
            agg, h, bg + (size_t)l * EMB, bnsum, bnsq, NNODES, 256);
        k_bn_params<<<1, EMB, 0, stream>>>(
            bnsum, bnsq, gamma + (size_t)l * EMB, beta + (size_t)l * EMB, scale, shift, NNODES);
        // h = act(t * scale + shift); no ReLU on last layer
        k_bn_apply<<<vec_blocks, 256, 0, stream>>>(
            (const float4*)agg, scale, shift, (float4*)h, NH4, (l == NLAYERS - 1) ? 0 : 1);
    }

    // ---- global_add_pool ----
    hipMemsetAsync(mol, 0, (size_t)NGRAPHS * EMB * sizeof(float), stream);
    k_pool<<<(NNODES + 255) / 256, EMB, 0, stream>>>(h, batch, mol, NNODES, 256);

    // ---- readout MLP ----
    k_gemm_wmma<EMB, EMB, 4><<<(NGRAPHS + 63) / 64, 256, 0, stream>>>(
        mol, W1, b1, m1, NGRAPHS, 1);
    k_gemm_wmma<EMB, 64, 4><<<(NGRAPHS + 63) / 64, 128, 0, stream>>>(
        m1, W2, b2, m2, NGRAPHS, 1);
    k_mlp3<<<(NGRAPHS * 32 + 255) / 256, 256, 0, stream>>>(m2, W3, b3, out, NGRAPHS);
}